// GNN_29892972380761
// MI455X (gfx1250) — compile-verified
//
#include <hip/hip_runtime.h>
#include <math.h>

#define N_NODES 100000
#define N_EDGES 3200000
#define HID 64

typedef float v2f __attribute__((ext_vector_type(2)));
typedef float v8f __attribute__((ext_vector_type(8)));

// ---------------------------------------------------------------- input layer
__global__ __launch_bounds__(256) void k_input(const float* __restrict__ x,
                                               const float* __restrict__ w_in,
                                               const float* __restrict__ b_in,
                                               const float* __restrict__ g_in,
                                               const float* __restrict__ bt_in,
                                               float* __restrict__ h) {
  int wave = threadIdx.x >> 5, lane = threadIdx.x & 31;
  int node = blockIdx.x * 8 + wave;
  if (node >= N_NODES) return;
  float x0 = x[node * 3 + 0], x1 = x[node * 3 + 1], x2 = x[node * 3 + 2];
  int c0 = lane, c1 = lane + 32;
  float h0 = b_in[c0] + x0 * w_in[c0] + x1 * w_in[64 + c0] + x2 * w_in[128 + c0];
  float h1 = b_in[c1] + x0 * w_in[c1] + x1 * w_in[64 + c1] + x2 * w_in[128 + c1];
  float s = h0 + h1, s2 = h0 * h0 + h1 * h1;
#pragma unroll
  for (int m = 1; m < 32; m <<= 1) {
    s += __shfl_xor(s, m, 32);
    s2 += __shfl_xor(s2, m, 32);
  }
  float mean = s * (1.f / 64.f);
  float var = s2 * (1.f / 64.f) - mean * mean;
  float rstd = rsqrtf(var + 1e-5f);
  h[node * 64 + c0] = tanhf((h0 - mean) * rstd * g_in[c0] + bt_in[c0]);
  h[node * 64 + c1] = tanhf((h1 - mean) * rstd * g_in[c1] + bt_in[c1]);
}

// ---------------------------------------------------------------- zero scratch
__global__ __launch_bounds__(256) void k_zero4(float4* __restrict__ p, int n4) {
  int i = blockIdx.x * 256 + threadIdx.x;
  if (i < n4) p[i] = make_float4(0.f, 0.f, 0.f, 0.f);
}

// ---------------------------------------------------------------- edge phase
// thread t -> (edge = t/16, 4-col chunk = (t%16)*4). float4 gather from L2,
// fp32 atomic scatter-add (global_atomic_add_f32, no return).
__global__ __launch_bounds__(256) void k_edges(const float* __restrict__ h,
                                               const int* __restrict__ esrc,
                                               const int* __restrict__ edst,
                                               const float* __restrict__ ev,
                                               float* __restrict__ mi,
                                               float* __restrict__ mo) {
  int t = blockIdx.x * 256 + threadIdx.x;
  int edge = t >> 4;
  if (edge >= N_EDGES) return;
  int c = (t & 15) * 4;
  int s = esrc[edge];
  int d = edst[edge];
  float w = ev[edge];
  float4 hs = *(const float4*)&h[s * 64 + c];
  float4 hd = *(const float4*)&h[d * 64 + c];
  atomicAdd(&mi[d * 64 + c + 0], w * hs.x);
  atomicAdd(&mi[d * 64 + c + 1], w * hs.y);
  atomicAdd(&mi[d * 64 + c + 2], w * hs.z);
  atomicAdd(&mi[d * 64 + c + 3], w * hs.w);
  atomicAdd(&mo[s * 64 + c + 0], w * hd.x);
  atomicAdd(&mo[s * 64 + c + 1], w * hd.y);
  atomicAdd(&mo[s * 64 + c + 2], w * hd.z);
  atomicAdd(&mo[s * 64 + c + 3], w * hd.w);
}

// ---------------------------------------------------------------- WMMA MLP
// One layer: D[16x64] = A[16xK] * W[Kx64] + b, then LN + tanh.
// 4 waves, wave w owns 16x16 tile of columns [16w,16w+16).
// A-fragment (f32 16x4): lanes 0-15 hold K=4k..4k+1, lanes 16-31 K=4k+2..4k+3.
__device__ __forceinline__ void wmma_layer(const float* __restrict__ Asrc, int astride,
                                           int ksteps, const float* __restrict__ W,
                                           const float* __restrict__ b,
                                           const float* __restrict__ g,
                                           const float* __restrict__ bt,
                                           float* act_pre, float* act_out, int tid) {
  const int wave = tid >> 5, lane = tid & 31;
  const int lr = lane & 15, hi = lane >> 4;
  const int colN = wave * 16 + lr;
  v8f acc = {};
  for (int k = 0; k < ksteps; ++k) {
    v2f a = *(const v2f*)&Asrc[lr * astride + 4 * k + 2 * hi];
    const float* wp = W + (4 * k + 2 * hi) * 64 + colN;
    v2f bv;
    bv.x = wp[0];
    bv.y = wp[64];
    acc = __builtin_amdgcn_wmma_f32_16x16x4_f32(false, a, false, bv, (short)0, acc,
                                                false, false);
  }
  float bias = b[colN];
#pragma unroll
  for (int v = 0; v < 8; ++v) act_pre[(v + 8 * hi) * 68 + colN] = acc[v] + bias;
  __syncthreads();
  // LayerNorm + tanh: 16 rows x 8 lane-groups, butterfly reduce within group of 8
  int row = tid >> 3, kk = tid & 7;
  const float* src = &act_pre[row * 68 + kk * 8];
  float vals[8];
  float s = 0.f, s2 = 0.f;
#pragma unroll
  for (int i = 0; i < 8; ++i) {
    float xv = src[i];
    vals[i] = xv;
    s += xv;
    s2 += xv * xv;
  }
#pragma unroll
  for (int m = 1; m < 8; m <<= 1) {
    s += __shfl_xor(s, m, 32);
    s2 += __shfl_xor(s2, m, 32);
  }
  float mean = s * (1.f / 64.f);
  float var = s2 * (1.f / 64.f) - mean * mean;
  float rstd = rsqrtf(var + 1e-5f);
#pragma unroll
  for (int i = 0; i < 8; ++i) {
    int c = kk * 8 + i;
    float y = (vals[i] - mean) * rstd * g[c] + bt[c];
    act_out[row * 68 + c] = tanhf(y);
  }
  __syncthreads();
}

__global__ __launch_bounds__(128) void k_mlp(
    const float* __restrict__ mi, const float* __restrict__ mo, float* __restrict__ h,
    const float* __restrict__ nw0, const float* __restrict__ nb0,
    const float* __restrict__ ng0, const float* __restrict__ nbt0,
    const float* __restrict__ nw1, const float* __restrict__ nb1,
    const float* __restrict__ ng1, const float* __restrict__ nbt1,
    const float* __restrict__ nw2, const float* __restrict__ nb2,
    const float* __restrict__ ng2, const float* __restrict__ nbt2,
    const float* __restrict__ nw3, const float* __restrict__ nb3,
    const float* __restrict__ ng3, const float* __restrict__ nbt3) {
  __shared__ float zA[16 * 196];    // 16 nodes x 192 (stride 196: bank rotation)
  __shared__ float act0[16 * 68];   // activations (layer inputs/outputs)
  __shared__ float act1[16 * 68];   // pre-LN scratch
  const int tid = threadIdx.x;
  const int nb = blockIdx.x * 16;  // 100000 = 6250 * 16, no tail

  // stage z = [mi | mo | h] as float4s
  for (int i = tid; i < 16 * 48; i += 128) {
    int row = i / 48, q = i % 48;
    int col = q * 4;
    int node = nb + row;
    const float* src = (col < 64)    ? (mi + node * 64 + col)
                       : (col < 128) ? (mo + node * 64 + (col - 64))
                                     : (h + node * 64 + (col - 128));
    *(float4*)&zA[row * 196 + col] = *(const float4*)src;
  }
  __syncthreads();

  wmma_layer(zA, 196, 48, nw0, nb0, ng0, nbt0, act1, act0, tid);
  wmma_layer(act0, 68, 16, nw1, nb1, ng1, nbt1, act1, act0, tid);
  wmma_layer(act0, 68, 16, nw2, nb2, ng2, nbt2, act1, act0, tid);
  wmma_layer(act0, 68, 16, nw3, nb3, ng3, nbt3, act1, act0, tid);

  // residual: h = z + h0 (in place; block owns its 16 nodes exclusively)
  for (int i = tid; i < 16 * 64; i += 128) {
    int row = i >> 6, col = i & 63;
    h[(nb + row) * 64 + col] += act0[row * 68 + col];
  }
}

// ---------------------------------------------------------------- segment max
__device__ __forceinline__ unsigned f2key(float f) {
  unsigned u = __float_as_uint(f);
  return (u & 0x80000000u) ? ~u : (u | 0x80000000u);
}
__device__ __forceinline__ float key2f(unsigned k) {
  return (k & 0x80000000u) ? __uint_as_float(k ^ 0x80000000u) : __uint_as_float(~k);
}

__global__ __launch_bounds__(256) void k_initkeys(unsigned* __restrict__ keys) {
  int i = blockIdx.x * 256 + threadIdx.x;
  if (i < 64 * 64) keys[i] = f2key(-INFINITY);
}

__global__ __launch_bounds__(256) void k_segmax(const float* __restrict__ h,
                                                const int* __restrict__ batch,
                                                unsigned* __restrict__ keys) {
  int t = blockIdx.x * 256 + threadIdx.x;
  if (t >= N_NODES * 64) return;
  int node = t >> 6, c = t & 63;
  int g = batch[node];
  atomicMax(&keys[g * 64 + c], f2key(h[t]));
}

// ---------------------------------------------------------------- readout
__global__ __launch_bounds__(64) void k_readout(const unsigned* __restrict__ keys,
                                                const float* __restrict__ ow0,
                                                const float* __restrict__ ob0,
                                                const float* __restrict__ og0,
                                                const float* __restrict__ obt0,
                                                const float* __restrict__ ow1,
                                                const float* __restrict__ ob1,
                                                float* __restrict__ out) {
  int g = threadIdx.x;  // 64 graphs, one thread each
  float s[64], t[64];
  for (int j = 0; j < 64; ++j) s[j] = key2f(keys[g * 64 + j]);
  float mean = 0.f;
  for (int j = 0; j < 64; ++j) {
    float acc = ob0[j];
    for (int k = 0; k < 64; ++k) acc += s[k] * ow0[k * 64 + j];
    t[j] = acc;
    mean += acc;
  }
  mean *= (1.f / 64.f);
  float var = 0.f;
  for (int j = 0; j < 64; ++j) {
    float d = t[j] - mean;
    var += d * d;
  }
  var *= (1.f / 64.f);
  float rstd = rsqrtf(var + 1e-5f);
  float o = ob1[0];
  for (int j = 0; j < 64; ++j)
    o += tanhf((t[j] - mean) * rstd * og0[j] + obt0[j]) * ow1[j];
  out[g] = o;
}

// ---------------------------------------------------------------- launcher
extern "C" void kernel_launch(void* const* d_in, const int* in_sizes, int n_in,
                              void* d_out, int out_size, void* d_ws, size_t ws_size,
                              hipStream_t stream) {
  const float* x = (const float*)d_in[0];
  const int* ei = (const int*)d_in[1];
  const int* batch = (const int*)d_in[2];
  const float* e = (const float*)d_in[4];
  const float* w_in = (const float*)d_in[5];
  const float* b_in = (const float*)d_in[6];
  const float* g_in = (const float*)d_in[7];
  const float* bt_in = (const float*)d_in[8];
  const float* nw[4], *nbv[4], *ng[4], *nbt[4];
  for (int i = 0; i < 4; ++i) {
    nw[i] = (const float*)d_in[9 + 4 * i];
    nbv[i] = (const float*)d_in[10 + 4 * i];
    ng[i] = (const float*)d_in[11 + 4 * i];
    nbt[i] = (const float*)d_in[12 + 4 * i];
  }
  const float* ow0 = (const float*)d_in[25];
  const float* ob0 = (const float*)d_in[26];
  const float* og0 = (const float*)d_in[27];
  const float* obt0 = (const float*)d_in[28];
  const float* ow1 = (const float*)d_in[29];
  const float* ob1 = (const float*)d_in[30];

  const size_t H = (size_t)N_NODES * 64;  // 6.4M floats
  float* ws = (float*)d_ws;
  float* h = ws;
  float* mi = ws + H;
  float* mo = ws + 2 * H;
  unsigned* keys = (unsigned*)(ws + 3 * H);

  k_input<<<(N_NODES + 7) / 8, 256, 0, stream>>>(x, w_in, b_in, g_in, bt_in, h);

  for (int it = 0; it < 3; ++it) {
    int n4 = (int)(2 * H / 4);
    k_zero4<<<(n4 + 255) / 256, 256, 0, stream>>>((float4*)mi, n4);
    k_edges<<<(N_EDGES * 16) / 256, 256, 0, stream>>>(h, ei, ei + N_EDGES, e, mi, mo);
    k_mlp<<<N_NODES / 16, 128, 0, stream>>>(
        mi, mo, h, nw[0], nbv[0], ng[0], nbt[0], nw[1], nbv[1], ng[1], nbt[1], nw[2],
        nbv[2], ng[2], nbt[2], nw[3], nbv[3], ng[3], nbt[3]);
  }

  k_initkeys<<<16, 256, 0, stream>>>(keys);
  k_segmax<<<(N_NODES * 64 + 255) / 256, 256, 0, stream>>>(h, batch, keys);
  k_readout<<<1, 64, 0, stream>>>(keys, ow0, ob0, og0, obt0, ow1, ob1, (float*)d_out);
}